// PNA_61280593379569
// MI455X (gfx1250) — compile-verified
//
#include <hip/hip_runtime.h>
#include <math.h>

// ---------------------------------------------------------------------------
// PNA GNN forward for MI455X (gfx1250, wave32, WMMA).
//   N=50000 nodes, E=800000 edges, G=128 graphs, H=64, L=2 conv layers.
// GEMMs run on V_WMMA_F32_16X16X4_F32 (fp32 fidelity, K=4 steps).
// Edge aggregation uses native f32 add atomics + order-encoded u32 min/max.
// ---------------------------------------------------------------------------

#define NODES   50000
#define EDGES   800000
#define GRAPHS  128
#define HDIM    64
#define CIN     192      // 3*H message width
#define AGG4    768      // 4 aggregators * 192
#define NTILES_N  (NODES/16)   // 3125 (exact)
#define NTILES_E  (EDGES/16)   // 50000 (exact)

typedef __attribute__((ext_vector_type(2))) float v2f;
typedef __attribute__((ext_vector_type(8))) float v8f;

static __device__ __forceinline__ v8f wmma4(v2f a, v2f b, v8f c) {
  // D = A(16x4,f32) * B(4x16,f32) + C(16x16,f32)
  return __builtin_amdgcn_wmma_f32_16x16x4_f32(false, a, false, b, (short)0, c,
                                               false, false);
}

// Order-preserving float<->uint mapping so integer min/max atomics give
// IEEE float min/max ordering.
static __device__ __forceinline__ unsigned ord_enc(float f) {
  unsigned u = __float_as_uint(f);
  return (u & 0x80000000u) ? ~u : (u | 0x80000000u);
}
static __device__ __forceinline__ float ord_dec(unsigned u) {
  unsigned b = (u & 0x80000000u) ? (u & 0x7fffffffu) : ~u;
  return __uint_as_float(b);
}

// ---------------------------------------------------------------------------
// Generic encoder GEMM: out[rows,64] = A[rows,K] @ W[K,64] + bias.
// One wave per 16-row tile, 4 column tiles of 16, K stepped by 4.
// A frag: lane L -> M = L&15, K = k0 + 2*(L>>4) + j   (j = vgpr 0/1)
// B frag: lane L -> N = L&15, K = k0 + 2*(L>>4) + j
// C frag: vgpr j, lane L -> M = j + 8*(L>>4), N = L&15
// ---------------------------------------------------------------------------
__global__ void enc_gemm_kernel(const float* __restrict__ A,
                                const float* __restrict__ W,
                                const float* __restrict__ bias,
                                float* __restrict__ out,
                                int K, int numTiles) {
  const int tile = blockIdx.x * (blockDim.x >> 5) + (threadIdx.x >> 5);
  if (tile >= numTiles) return;                 // whole-wave uniform exit
  const int lane = threadIdx.x & 31;
  const int half = lane >> 4;
  const int l16  = lane & 15;
  const long m0  = (long)tile * 16;

  v8f c[4];
  #pragma unroll
  for (int nt = 0; nt < 4; ++nt)
    #pragma unroll
    for (int j = 0; j < 8; ++j) c[nt][j] = 0.f;

  const long arow = (m0 + l16) * (long)K;
  for (int k0 = 0; k0 < K; k0 += 4) {
    v2f a;
    a[0] = A[arow + k0 + 2 * half + 0];
    a[1] = A[arow + k0 + 2 * half + 1];
    const int wr = (k0 + 2 * half) * HDIM;
    #pragma unroll
    for (int nt = 0; nt < 4; ++nt) {
      v2f b;
      b[0] = W[wr + nt * 16 + l16];
      b[1] = W[wr + HDIM + nt * 16 + l16];
      c[nt] = wmma4(a, b, c[nt]);
    }
  }

  #pragma unroll
  for (int nt = 0; nt < 4; ++nt) {
    const int col = nt * 16 + l16;
    const float bv = bias[col];
    #pragma unroll
    for (int j = 0; j < 8; ++j) {
      const long row = m0 + j + 8 * half;
      out[row * HDIM + col] = c[nt][j] + bv;
    }
  }
}

// ---------------------------------------------------------------------------
// PNA conv GEMM with fused degree scalers:
//   h = agg@W0 + amp*(agg@W1) + att*(agg@W2) + bias
// agg: [N,768] = [mean|min|max|std]; W: [2304,64] (rows 0..767 = W0, etc.)
// ---------------------------------------------------------------------------
__global__ void conv_gemm_kernel(const float* __restrict__ agg,
                                 const float* __restrict__ W,
                                 const float* __restrict__ bias,
                                 const float* __restrict__ amp,
                                 const float* __restrict__ att,
                                 float* __restrict__ h,
                                 int numTiles) {
  const int tile = blockIdx.x * (blockDim.x >> 5) + (threadIdx.x >> 5);
  if (tile >= numTiles) return;
  const int lane = threadIdx.x & 31;
  const int half = lane >> 4;
  const int l16  = lane & 15;
  const long m0  = (long)tile * 16;

  v8f c0[4], c1[4], c2[4];
  #pragma unroll
  for (int nt = 0; nt < 4; ++nt)
    #pragma unroll
    for (int j = 0; j < 8; ++j) { c0[nt][j] = 0.f; c1[nt][j] = 0.f; c2[nt][j] = 0.f; }

  const float* __restrict__ W1 = W + (long)AGG4 * HDIM;
  const float* __restrict__ W2 = W + 2L * AGG4 * HDIM;
  const long arow = (m0 + l16) * (long)AGG4;

  for (int k0 = 0; k0 < AGG4; k0 += 4) {
    v2f a;
    a[0] = agg[arow + k0 + 2 * half + 0];
    a[1] = agg[arow + k0 + 2 * half + 1];
    __builtin_prefetch(&agg[arow + k0 + 64], 0, 1);   // global_prefetch_b8
    const int wr = (k0 + 2 * half) * HDIM;
    #pragma unroll
    for (int nt = 0; nt < 4; ++nt) {
      const int cidx = wr + nt * 16 + l16;
      v2f b;
      b[0] = W[cidx];  b[1] = W[cidx + HDIM];
      c0[nt] = wmma4(a, b, c0[nt]);
      b[0] = W1[cidx]; b[1] = W1[cidx + HDIM];
      c1[nt] = wmma4(a, b, c1[nt]);
      b[0] = W2[cidx]; b[1] = W2[cidx + HDIM];
      c2[nt] = wmma4(a, b, c2[nt]);
    }
  }

  #pragma unroll
  for (int j = 0; j < 8; ++j) {
    const long row = m0 + j + 8 * half;
    const float am = amp[row];
    const float at = att[row];
    #pragma unroll
    for (int nt = 0; nt < 4; ++nt) {
      const int col = nt * 16 + l16;
      h[row * HDIM + col] = c0[nt][j] + am * c1[nt][j] + at * c2[nt][j] + bias[col];
    }
  }
}

// ------------------------------ utility kernels -----------------------------
__global__ void zero_f32_kernel(float* __restrict__ p, long n) {
  long i = (long)blockIdx.x * blockDim.x + threadIdx.x;
  const long s = (long)gridDim.x * blockDim.x;
  for (; i < n; i += s) p[i] = 0.f;
}

// Accumulator layout per node: [0,192)=sum [192,384)=sumsq
//                              [384,576)=min(ord-enc) [576,768)=max(ord-enc)
__global__ void init_acc_kernel(unsigned* __restrict__ acc, long n) {
  long i = (long)blockIdx.x * blockDim.x + threadIdx.x;
  const long s = (long)gridDim.x * blockDim.x;
  for (; i < n; i += s) {
    const int slot = (int)(i % AGG4);
    acc[i] = (slot < 384) ? 0u : (slot < 576 ? 0xFFFFFFFFu : 0u);
  }
}

__global__ void degree_kernel(const int* __restrict__ dst,
                              float* __restrict__ deg, long ne) {
  long i = (long)blockIdx.x * blockDim.x + threadIdx.x;
  const long s = (long)gridDim.x * blockDim.x;
  for (; i < ne; i += s) atomicAdd(&deg[dst[i]], 1.0f);
}

__global__ void logdeg_kernel(const float* __restrict__ deg,
                              float* __restrict__ stats, long n) {
  __shared__ float red[256];
  long i = (long)blockIdx.x * blockDim.x + threadIdx.x;
  const long s = (long)gridDim.x * blockDim.x;
  float acc = 0.f;
  for (; i < n; i += s) acc += logf(deg[i] + 1.0f);
  red[threadIdx.x] = acc;
  __syncthreads();
  for (int off = 128; off > 0; off >>= 1) {
    if ((int)threadIdx.x < off) red[threadIdx.x] += red[threadIdx.x + off];
    __syncthreads();
  }
  if (threadIdx.x == 0) atomicAdd(&stats[0], red[0]);
}

__global__ void amp_att_kernel(const float* __restrict__ deg,
                               const float* __restrict__ stats,
                               float* __restrict__ amp,
                               float* __restrict__ att, long n) {
  long i = (long)blockIdx.x * blockDim.x + threadIdx.x;
  const long s = (long)gridDim.x * blockDim.x;
  const float avg = stats[0] / (float)NODES;
  for (; i < n; i += s) {
    const float ld = logf(deg[i] + 1.0f);
    amp[i] = ld / avg;
    att[i] = avg / fmaxf(ld, 0.693147180559945f);   // log(2)
  }
}

// Edge scatter: message h_e = [z[dst] | z[src] | ea[e]] (192 feats), reduce
// sum / sumsq / min / max into acc[dst].
__global__ void scatter_kernel(const float* __restrict__ z,
                               const float* __restrict__ ea,
                               const int* __restrict__ src,
                               const int* __restrict__ dst,
                               float* __restrict__ acc, long total) {
  long i = (long)blockIdx.x * blockDim.x + threadIdx.x;
  const long s = (long)gridDim.x * blockDim.x;
  for (; i < total; i += s) {
    const long e = i / CIN;
    const int  f = (int)(i % CIN);
    const int  d = dst[e];
    float v;
    if (f < 64)       v = z[(long)d * HDIM + f];
    else if (f < 128) v = z[(long)src[e] * HDIM + (f - 64)];
    else              v = ea[e * HDIM + (f - 128)];
    float* base = acc + (long)d * AGG4;
    atomicAdd(base + f, v);
    atomicAdd(base + CIN + f, v * v);
    const unsigned enc = ord_enc(v);
    atomicMin((unsigned*)(base + 2 * CIN) + f, enc);
    atomicMax((unsigned*)(base + 3 * CIN) + f, enc);
  }
}

// In-place transform acc -> agg = [mean | min | max | std]
__global__ void finalize_kernel(float* __restrict__ acc,
                                const float* __restrict__ deg, long total) {
  long i = (long)blockIdx.x * blockDim.x + threadIdx.x;
  const long s = (long)gridDim.x * blockDim.x;
  for (; i < total; i += s) {
    const long n = i / CIN;
    const int  f = (int)(i % CIN);
    float* b = acc + n * AGG4;
    const float  sum = b[f];
    const float  sq  = b[CIN + f];
    const unsigned mnb = ((unsigned*)b)[2 * CIN + f];
    const unsigned mxb = ((unsigned*)b)[3 * CIN + f];
    const float dv = deg[n];
    const float d  = fmaxf(dv, 1.0f);
    const float mean = sum / d;
    const float var  = fmaxf(sq / d - mean * mean, 0.0f);
    const float sd   = sqrtf(var + 1e-5f);
    const float mn   = (dv > 0.f) ? ord_dec(mnb) : 0.f;
    const float mx   = (dv > 0.f) ? ord_dec(mxb) : 0.f;
    b[f]           = mean;
    b[CIN + f]     = mn;
    b[2 * CIN + f] = mx;
    b[3 * CIN + f] = sd;
  }
}

// BatchNorm column stats over N rows (stats[64..127]=sum, [128..191]=sumsq)
__global__ void bn_stats_kernel(const float* __restrict__ h,
                                float* __restrict__ stats, int rows) {
  const int tid  = blockIdx.x * blockDim.x + threadIdx.x;
  const int col  = tid & 63;
  const int r0   = tid >> 6;
  const int rstr = (gridDim.x * blockDim.x) >> 6;
  float s = 0.f, s2 = 0.f;
  for (int r = r0; r < rows; r += rstr) {
    const float v = h[(long)r * HDIM + col];
    s += v; s2 += v * v;
  }
  atomicAdd(&stats[64 + col], s);
  atomicAdd(&stats[128 + col], s2);
}

// z = relu(bn(h)) + z   (in place on z)
__global__ void bn_apply_kernel(const float* __restrict__ h,
                                float* __restrict__ z,
                                const float* __restrict__ stats,
                                const float* __restrict__ gamma,
                                const float* __restrict__ beta, long total) {
  long i = (long)blockIdx.x * blockDim.x + threadIdx.x;
  const long s = (long)gridDim.x * blockDim.x;
  const float invN = 1.0f / (float)NODES;
  for (; i < total; i += s) {
    const int col = (int)(i & 63);
    const float mu  = stats[64 + col] * invN;
    const float var = stats[128 + col] * invN - mu * mu;
    const float y = (h[i] - mu) * rsqrtf(var + 1e-5f) * gamma[col] + beta[col];
    z[i] += fmaxf(y, 0.f);
  }
}

// Global mean-pool accumulation
__global__ void pool_kernel(const float* __restrict__ z,
                            const int* __restrict__ batch,
                            float* __restrict__ g, float* __restrict__ cnt,
                            long total) {
  long i = (long)blockIdx.x * blockDim.x + threadIdx.x;
  const long s = (long)gridDim.x * blockDim.x;
  for (; i < total; i += s) {
    const long n = i >> 6;
    const int  f = (int)(i & 63);
    const int  b = batch[n];
    atomicAdd(&g[(long)b * HDIM + f], z[i]);
    if (f == 0) atomicAdd(&cnt[b], 1.0f);
  }
}

// Readout MLP: 128 graphs, 64 -> 32 -> 16 -> 2 (one thread per graph)
__global__ void head_kernel(const float* __restrict__ g,
                            const float* __restrict__ cnt,
                            const float* __restrict__ w1, const float* __restrict__ b1,
                            const float* __restrict__ w2, const float* __restrict__ b2,
                            const float* __restrict__ w3, const float* __restrict__ b3,
                            float* __restrict__ out) {
  __shared__ float sw1[64 * 32], sb1[32], sw2[32 * 16], sb2[16], sw3[16 * 2], sb3[2];
  for (int i = threadIdx.x; i < 64 * 32; i += blockDim.x) sw1[i] = w1[i];
  for (int i = threadIdx.x; i < 32;      i += blockDim.x) sb1[i] = b1[i];
  for (int i = threadIdx.x; i < 32 * 16; i += blockDim.x) sw2[i] = w2[i];
  for (int i = threadIdx.x; i < 16;      i += blockDim.x) sb2[i] = b2[i];
  for (int i = threadIdx.x; i < 16 * 2;  i += blockDim.x) sw3[i] = w3[i];
  for (int i = threadIdx.x; i < 2;       i += blockDim.x) sb3[i] = b3[i];
  __syncthreads();

  const int r = threadIdx.x;
  if (r >= GRAPHS) return;
  const float inv = 1.0f / fmaxf(cnt[r], 1.0f);
  float gi[64];
  for (int k = 0; k < 64; ++k) gi[k] = g[(long)r * HDIM + k] * inv;
  float h1[32];
  for (int o = 0; o < 32; ++o) {
    float s = sb1[o];
    for (int k = 0; k < 64; ++k) s += gi[k] * sw1[k * 32 + o];
    h1[o] = fmaxf(s, 0.f);
  }
  float h2[16];
  for (int o = 0; o < 16; ++o) {
    float s = sb2[o];
    for (int k = 0; k < 32; ++k) s += h1[k] * sw2[k * 16 + o];
    h2[o] = fmaxf(s, 0.f);
  }
  for (int o = 0; o < 2; ++o) {
    float s = sb3[o];
    for (int k = 0; k < 16; ++k) s += h2[k] * sw3[k * 2 + o];
    out[r * 2 + o] = s;
  }
}

// ---------------------------------------------------------------------------
// Host launcher. Input order (setup_inputs insertion order, params flattened
// in declaration order):
//  0:x 1:edge_attr
//  2:node_enc.w 3:node_enc.b 4:edge_enc.w 5:edge_enc.b
//  6:convs0.w 7:convs0.b 8:convs1.w 9:convs1.b
// 10:bn0.gamma 11:bn0.beta 12:bn1.gamma 13:bn1.beta
// 14:fc1.w 15:fc1.b 16:fc2.w 17:fc2.b 18:fc3.w 19:fc3.b
// 20:edge_index[2,E] 21:batch[N]
// ---------------------------------------------------------------------------
extern "C" void kernel_launch(void* const* d_in, const int* in_sizes, int n_in,
                              void* d_out, int out_size, void* d_ws, size_t ws_size,
                              hipStream_t stream) {
  (void)in_sizes; (void)n_in; (void)out_size; (void)ws_size;

  const float* x     = (const float*)d_in[0];
  const float* eattr = (const float*)d_in[1];
  const float* neW   = (const float*)d_in[2];
  const float* neB   = (const float*)d_in[3];
  const float* eeW   = (const float*)d_in[4];
  const float* eeB   = (const float*)d_in[5];
  const float* convW[2] = {(const float*)d_in[6], (const float*)d_in[8]};
  const float* convB[2] = {(const float*)d_in[7], (const float*)d_in[9]};
  const float* bnG[2]   = {(const float*)d_in[10], (const float*)d_in[12]};
  const float* bnB[2]   = {(const float*)d_in[11], (const float*)d_in[13]};
  const float* fc1w = (const float*)d_in[14];
  const float* fc1b = (const float*)d_in[15];
  const float* fc2w = (const float*)d_in[16];
  const float* fc2b = (const float*)d_in[17];
  const float* fc3w = (const float*)d_in[18];
  const float* fc3b = (const float*)d_in[19];
  const int*   eidx  = (const int*)d_in[20];
  const int*   src   = eidx;
  const int*   dst   = eidx + EDGES;
  const int*   batch = (const int*)d_in[21];
  float* out = (float*)d_out;

  // ---- workspace carve-up (floats) ----
  float* ws = (float*)d_ws;
  long o = 0;
  float* z     = ws + o; o += (long)NODES * HDIM;   // 3.2M node features
  float* h     = ws + o; o += (long)NODES * HDIM;   // 3.2M conv output
  float* ea    = ws + o; o += (long)EDGES * HDIM;   // 51.2M encoded edge attr
  float* acc   = ws + o; o += (long)NODES * AGG4;   // 38.4M accumulators/agg
  float* deg   = ws + o; o += NODES;
  float* amp   = ws + o; o += NODES;
  float* att   = ws + o; o += NODES;
  float* stats = ws + o; o += 256;                  // [0]=sum log_deg, [64..]=bn
  float* gpool = ws + o; o += (long)GRAPHS * HDIM;
  float* cnt   = ws + o; o += GRAPHS;               // contiguous after gpool

  // 1) encoders (WMMA f32 GEMMs)
  enc_gemm_kernel<<<(NTILES_N + 3) / 4, 128, 0, stream>>>(x, neW, neB, z, 64, NTILES_N);
  enc_gemm_kernel<<<(NTILES_E + 3) / 4, 128, 0, stream>>>(eattr, eeW, eeB, ea, 8, NTILES_E);

  // 2) degree + PNA scalers
  zero_f32_kernel<<<64, 256, 0, stream>>>(deg, NODES);
  zero_f32_kernel<<<1, 256, 0, stream>>>(stats, 256);
  degree_kernel<<<2048, 256, 0, stream>>>(dst, deg, EDGES);
  logdeg_kernel<<<256, 256, 0, stream>>>(deg, stats, NODES);
  amp_att_kernel<<<256, 256, 0, stream>>>(deg, stats, amp, att, NODES);

  // 3) two PNA conv layers
  for (int l = 0; l < 2; ++l) {
    init_acc_kernel<<<4096, 256, 0, stream>>>((unsigned*)acc, (long)NODES * AGG4);
    scatter_kernel<<<8192, 256, 0, stream>>>(z, ea, src, dst, acc, (long)EDGES * CIN);
    finalize_kernel<<<8192, 256, 0, stream>>>(acc, deg, (long)NODES * CIN);
    conv_gemm_kernel<<<(NTILES_N + 3) / 4, 128, 0, stream>>>(
        acc, convW[l], convB[l], amp, att, h, NTILES_N);
    zero_f32_kernel<<<1, 256, 0, stream>>>(stats + 64, 128);
    bn_stats_kernel<<<64, 256, 0, stream>>>(h, stats, NODES);
    bn_apply_kernel<<<4096, 256, 0, stream>>>(h, z, stats, bnG[l], bnB[l],
                                              (long)NODES * HDIM);
  }

  // 4) mean pool + readout MLP
  zero_f32_kernel<<<16, 256, 0, stream>>>(gpool, (long)GRAPHS * HDIM + GRAPHS);
  pool_kernel<<<4096, 256, 0, stream>>>(z, batch, gpool, cnt, (long)NODES * HDIM);
  head_kernel<<<1, 128, 0, stream>>>(gpool, cnt, fc1w, fc1b, fc2w, fc2b, fc3w, fc3b, out);
}